// GraphPairBaseline_21242908246054
// MI455X (gfx1250) — compile-verified
//
#include <hip/hip_runtime.h>

#define NN 100000
#define NE 6400000
#define HID 32
#define WPB 8   // waves per block in MLP kernel

typedef __attribute__((ext_vector_type(2))) float v2f;
typedef __attribute__((ext_vector_type(8))) float v8f;

static __device__ __forceinline__ v8f zero8() {
    v8f z;
#pragma unroll
    for (int i = 0; i < 8; ++i) z[i] = 0.0f;
    return z;
}

static __device__ __forceinline__ v8f wmma4(v2f a, v2f b, v8f c) {
    // D = A(16x4 f32) * B(4x16 f32) + C(16x16 f32)
    return __builtin_amdgcn_wmma_f32_16x16x4_f32(
        false, a, false, b, (short)0, c, false, false);
}

__global__ void zero_kernel(float* __restrict__ p, int n) {
    int i = blockIdx.x * blockDim.x + threadIdx.x;
    if (i < n) p[i] = 0.0f;
}

// layer-0 scatter: x is N x 1
__global__ void scatter_d1(const float* __restrict__ x,
                           const int* __restrict__ src,
                           const int* __restrict__ dst,
                           float* __restrict__ agg, int ne) {
    int e = blockIdx.x * blockDim.x + threadIdx.x;
    if (e < ne) atomicAdd(&agg[dst[e]], x[src[e]]);
}

// 32-dim scatter: lane = feature, contiguous 128B gather per edge; atomics land in L2
__global__ void scatter_d32(const float* __restrict__ x,
                            const int* __restrict__ src,
                            const int* __restrict__ dst,
                            float* __restrict__ agg, int ne) {
    int tid = blockIdx.x * blockDim.x + threadIdx.x;
    int e = tid >> 5;
    int f = tid & 31;
    if (e < ne) {
        int s = src[e], d = dst[e];
        atomicAdd(&agg[d * HID + f], x[s * HID + f]);
    }
}

// Fused GIN MLP for one layer:
//   h = xin + agg   (din = 1 or 32)
//   T = relu(h @ W1 + bias1)          [GEMM1 via WMMA when din==32]
//   xout = relu(T @ W2 + bias2)       [GEMM2 via WMMA, outer relu folded in]
// One wave per 16-row tile; T staged through LDS between GEMMs.
__global__ __launch_bounds__(256) void gin_mlp_kernel(
    const float* __restrict__ xin, const float* __restrict__ agg,
    const float* __restrict__ W1, const float* __restrict__ bias1,
    const float* __restrict__ W2, const float* __restrict__ bias2,
    float* __restrict__ xout, int n, int din) {
    __shared__ float ldsT[WPB][16 * 33];

    const int lane = threadIdx.x & 31;
    const int wave = threadIdx.x >> 5;
    const int half = lane >> 4;   // 0: lanes 0-15, 1: lanes 16-31
    const int l16 = lane & 15;
    const int tile = blockIdx.x * WPB + wave;
    const int row0 = tile * 16;
    const bool active = (row0 < n);
    float* T = ldsT[wave];

    if (active) {
        if (din == 1) {
            // T[r][c] = relu(h[r] * W1[0][c] + bias1[c]); lane owns column `lane`
            const float w = W1[lane];
            const float b = bias1[lane];
#pragma unroll 4
            for (int r = 0; r < 16; ++r) {
                float h = xin[row0 + r] + agg[row0 + r];
                T[r * 33 + lane] = fmaxf(h * w + b, 0.0f);
            }
        } else {
            // GEMM1: C(16x32) = A(16x32) * W1(32x32)
            const float* xr = xin + (row0 + l16) * HID;
            const float* ar = agg + (row0 + l16) * HID;
            v8f c0 = zero8();
            v8f c1 = zero8();
#pragma unroll
            for (int k0 = 0; k0 < HID; k0 += 4) {
                int ka = k0 + 2 * half;   // lane group selects K sub-pair
                v2f a;
                a[0] = xr[ka] + ar[ka];
                a[1] = xr[ka + 1] + ar[ka + 1];
                v2f b, bb;
                b[0]  = W1[ka * HID + l16];
                b[1]  = W1[(ka + 1) * HID + l16];
                bb[0] = W1[ka * HID + 16 + l16];
                bb[1] = W1[(ka + 1) * HID + 16 + l16];
                c0 = wmma4(a, b, c0);
                c1 = wmma4(a, bb, c1);
            }
            const float bA = bias1[l16];
            const float bB = bias1[16 + l16];
#pragma unroll
            for (int v = 0; v < 8; ++v) {
                int r = v + half * 8;   // C layout: VGPR v -> row v (+8 for upper lanes)
                T[r * 33 + l16]      = fmaxf(c0[v] + bA, 0.0f);
                T[r * 33 + 16 + l16] = fmaxf(c1[v] + bB, 0.0f);
            }
        }
    }
    __syncthreads();
    if (active) {
        // GEMM2: xout(16x32) = T(16x32) * W2(32x32)
        v8f c0 = zero8();
        v8f c1 = zero8();
#pragma unroll
        for (int k0 = 0; k0 < HID; k0 += 4) {
            int ka = k0 + 2 * half;
            v2f a;
            a[0] = T[l16 * 33 + ka];
            a[1] = T[l16 * 33 + ka + 1];
            v2f b, bb;
            b[0]  = W2[ka * HID + l16];
            b[1]  = W2[(ka + 1) * HID + l16];
            bb[0] = W2[ka * HID + 16 + l16];
            bb[1] = W2[(ka + 1) * HID + 16 + l16];
            c0 = wmma4(a, b, c0);
            c1 = wmma4(a, bb, c1);
        }
        const float bA = bias2[l16];
        const float bB = bias2[16 + l16];
#pragma unroll
        for (int v = 0; v < 8; ++v) {
            int r = row0 + v + half * 8;
            xout[r * HID + l16]      = fmaxf(c0[v] + bA, 0.0f);
            xout[r * HID + 16 + l16] = fmaxf(c1[v] + bB, 0.0f);
        }
    }
}

// global add pool: pooled[32] += sum over rows of x
__global__ __launch_bounds__(256) void pool_kernel(const float* __restrict__ x,
                                                   float* __restrict__ pooled, int n) {
    __shared__ float red[8][32];
    const int lane = threadIdx.x & 31;
    const int wave = threadIdx.x >> 5;
    float acc = 0.0f;
    for (int r = blockIdx.x * 8 + wave; r < n; r += gridDim.x * 8)
        acc += x[r * HID + lane];
    red[wave][lane] = acc;
    __syncthreads();
    if (wave == 0) {
        float s = red[0][lane];
#pragma unroll
        for (int w = 1; w < 8; ++w) s += red[w][lane];
        atomicAdd(&pooled[lane], s);
    }
}

// final combine MLP + sigmoid (single wave)
__global__ void combine_kernel(const float* __restrict__ pg, const float* __restrict__ ph,
                               const float* __restrict__ Wc1, const float* __restrict__ bc1,
                               const float* __restrict__ Wc2, const float* __restrict__ bc2,
                               float* __restrict__ out) {
    int j = threadIdx.x;   // 0..31
    float u = bc1[j];
    for (int k = 0; k < HID; ++k) u += pg[k] * Wc1[k * HID + j];
    for (int k = 0; k < HID; ++k) u += ph[k] * Wc1[(HID + k) * HID + j];
    u = fmaxf(u, 0.0f);
    float p = u * Wc2[j];
#pragma unroll
    for (int off = 16; off > 0; off >>= 1) p += __shfl_down(p, off, 32);
    if (j == 0) out[0] = 1.0f / (1.0f + __expf(-(p + bc2[0])));
}

extern "C" void kernel_launch(void* const* d_in, const int* in_sizes, int n_in,
                              void* d_out, int out_size, void* d_ws, size_t ws_size,
                              hipStream_t stream) {
    const float* xG  = (const float*)d_in[0];
    const int*   eiG = (const int*)d_in[1];
    const float* xH  = (const float*)d_in[2];
    const int*   eiH = (const int*)d_in[3];
    const float* W1[3]  = {(const float*)d_in[4],  (const float*)d_in[8],  (const float*)d_in[12]};
    const float* b1_[3] = {(const float*)d_in[5],  (const float*)d_in[9],  (const float*)d_in[13]};
    const float* W2[3]  = {(const float*)d_in[6],  (const float*)d_in[10], (const float*)d_in[14]};
    const float* b2_[3] = {(const float*)d_in[7],  (const float*)d_in[11], (const float*)d_in[15]};
    const float* Wc1 = (const float*)d_in[16];
    const float* bc1 = (const float*)d_in[17];
    const float* Wc2 = (const float*)d_in[18];
    const float* bc2 = (const float*)d_in[19];

    float* bufX    = (float*)d_ws;          // N x 32
    float* bufAgg  = bufX + NN * HID;       // N x 32
    float* pooledG = bufAgg + NN * HID;     // 32
    float* pooledH = pooledG + HID;         // 32

    const int numTiles = (NN + 15) / 16;
    const dim3 mlpGrid((numTiles + WPB - 1) / WPB);
    const int zeroN32 = (NN * HID + 255) / 256;
    const int zeroN1  = (NN + 255) / 256;
    const int sc1Grid = (NE + 255) / 256;
    const int sc32Grid = (int)(((long long)NE * 32 + 255) / 256);

    // zero pooled sums (adjacent)
    zero_kernel<<<1, 64, 0, stream>>>(pooledG, 64);

    const float* xNode[2]  = {xG, xH};
    const int*   eidx[2]   = {eiG, eiH};
    float*       pooled[2] = {pooledG, pooledH};

    for (int g = 0; g < 2; ++g) {
        const int* src = eidx[g];
        const int* dst = eidx[g] + NE;
        // layer 0 (din = 1)
        zero_kernel<<<zeroN1, 256, 0, stream>>>(bufAgg, NN);
        scatter_d1<<<sc1Grid, 256, 0, stream>>>(xNode[g], src, dst, bufAgg, NE);
        gin_mlp_kernel<<<mlpGrid, 256, 0, stream>>>(xNode[g], bufAgg,
                                                    W1[0], b1_[0], W2[0], b2_[0],
                                                    bufX, NN, 1);
        // layers 1, 2 (din = 32)
        for (int l = 1; l < 3; ++l) {
            zero_kernel<<<zeroN32, 256, 0, stream>>>(bufAgg, NN * HID);
            scatter_d32<<<sc32Grid, 256, 0, stream>>>(bufX, src, dst, bufAgg, NE);
            gin_mlp_kernel<<<mlpGrid, 256, 0, stream>>>(bufX, bufAgg,
                                                        W1[l], b1_[l], W2[l], b2_[l],
                                                        bufX, NN, HID);
        }
        pool_kernel<<<512, 256, 0, stream>>>(bufX, pooled[g], NN);
    }

    combine_kernel<<<1, 32, 0, stream>>>(pooledG, pooledH, Wc1, bc1, Wc2, bc2,
                                         (float*)d_out);
}